// WildfireModel_75917841924540
// MI455X (gfx1250) — compile-verified
//
#include <hip/hip_runtime.h>
#include <stdint.h>

#define HH 2048
#define WW 2048
#define TX 64
#define TY 16
#define LDSW 68   // (TX+2) rounded up to a multiple of 4 bytes for the TDM tile
#define LDSH 18   // TY+2

typedef __attribute__((ext_vector_type(4))) unsigned int su4;
typedef __attribute__((ext_vector_type(8))) unsigned int su8;

__launch_bounds__(256)
__global__ void wildfire_step_kernel(
    const float* __restrict__ p_veg, const float* __restrict__ p_den,
    const float* __restrict__ wv,    const float* __restrict__ wdir,
    const float* __restrict__ slope,
    const float* __restrict__ a_p,   const float* __restrict__ ph_p,
    const float* __restrict__ c1_p,  const float* __restrict__ c2_p,
    const float* __restrict__ pc_p,
    const float* __restrict__ rand_ignite, const float* __restrict__ rand_continue,
    const unsigned char* __restrict__ burning, const unsigned char* __restrict__ burned,
    float* __restrict__ out0, float* __restrict__ out1, float* __restrict__ out2)
{
    __shared__ __align__(16) unsigned char sb[LDSH * LDSW];

    const int tid = threadIdx.x;
    const int bx = blockIdx.x, by = blockIdx.y;
    const int gx0 = bx * TX, gy0 = by * TY;

    const bool interior = (bx > 0) & (bx < (WW / TX - 1)) & (by > 0) & (by < (HH / TY - 1));

    if (interior) {
        // Interior tile (incl. halo) is fully in-bounds: stage it with one TDM op.
        // TDM ignores EXEC and issues once per wave executing it -> wave 0 only.
        if (tid < 32) {
            const uint64_t gaddr =
                (uint64_t)(uintptr_t)(burning + (size_t)(gy0 - 1) * WW + (gx0 - 1));
            const uint32_t lds = (uint32_t)(uintptr_t)(&sb[0]); // low 32 bits = LDS offset

            su4 g0; su8 g1; su4 gz;
            // Group 0: count=1 | lds_addr | global_addr[56:0] | type=2 (image)
            g0.x = 1u;
            g0.y = lds;
            g0.z = (uint32_t)gaddr;
            g0.w = (uint32_t)((gaddr >> 32) & 0x01FFFFFFu) | (2u << 30);
            // Group 1: mask=0, data_size=1B, no pad; tensor 2048x2048, tile 68x18, stride 2048
            g1.s0 = 0u;
            g1.s1 = ((uint32_t)(WW & 0xFFFF)) << 16;                          // tensor_dim0 lo16
            g1.s2 = (uint32_t)(WW >> 16) | (((uint32_t)(HH & 0xFFFF)) << 16); // dim0 hi | dim1 lo
            g1.s3 = (uint32_t)(HH >> 16) | (((uint32_t)LDSW) << 16);          // dim1 hi | tile_dim0
            g1.s4 = (uint32_t)LDSH;                                           // tile_dim1, tile_dim2=0
            g1.s5 = (uint32_t)WW;                                             // tensor_dim0_stride lo
            g1.s6 = 0u;
            g1.s7 = 0u;
            gz.x = 0u; gz.y = 0u; gz.z = 0u; gz.w = 0u;                       // unused dims 2..4

            asm volatile("tensor_load_to_lds %0, %1, %2, %3"
                         :
                         : "s"(g0), "s"(g1), "s"(gz), "s"(gz)
                         : "memory");
            __builtin_amdgcn_s_wait_tensorcnt(0);
        }
    } else {
        // Edge tiles: guarded cooperative halo load (zero padding semantics).
        for (int i = tid; i < LDSH * LDSW; i += 256) {
            const int r = i / LDSW, c = i - r * LDSW;
            const int gy = gy0 - 1 + r, gx = gx0 - 1 + c;
            unsigned char v = 0;
            if ((unsigned)gy < (unsigned)HH && (unsigned)gx < (unsigned)WW)
                v = burning[(size_t)gy * WW + gx];
            sb[i] = v;
        }
    }
    __syncthreads();

    // Uniform scalar parameters
    const float a  = *a_p;
    const float ph = *ph_p;
    const float c1 = *c1_p;
    const float c2 = *c2_p;
    const float pc = *pc_p;
    const float RAD    = 0.017453292519943295f;
    const float TANH_C = 1.1486328125f;
    const float aR     = a * RAD;
    const float R2     = 0.70710678118654752f;
    // DIR_DEG = [315,270,225, 0,-,180, 45,90,135] flattened row-major
    const float DCOS[9] = {  R2, 0.f, -R2, 1.f, 0.f, -1.f, R2, 0.f, -R2 };
    const float DSIN[9] = { -R2, -1.f, -R2, 0.f, 0.f, 0.f, R2, 1.f,  R2 };

    // Thread -> 4 consecutive columns of one row. 16 x-threads * 16 y-threads.
    const int txi = (tid & 15) * 4;   // starting column within tile (mult of 4)
    const int ty  = tid >> 4;         // row within tile (0..15)
    const int gx  = gx0 + txi, gy = gy0 + ty;
    const size_t g = (size_t)gy * WW + gx;

    // 3 halo rows, 2 aligned words each = bytes [txi .. txi+7] of each halo row,
    // i.e. cells [txi-1 .. txi+6] -> covers neighborhoods of cells txi..txi+3.
    uint64_t row[3];
#pragma unroll
    for (int r = 0; r < 3; ++r) {
        const int base = (ty + r) * LDSW + txi;
        const uint32_t w0 = *(const uint32_t*)(sb + base);
        const uint32_t w1 = *(const uint32_t*)(sb + base + 4);
        row[r] = (uint64_t)w0 | ((uint64_t)w1 << 32);
    }
    const uint32_t burned4 = *(const uint32_t*)(burned + g);

    float4 pig, nbn, nbd;
    float*       pigf = &pig.x;
    float*       nbnf = &nbn.x;
    float*       nbdf = &nbd.x;

#pragma unroll
    for (int i = 0; i < 4; ++i) {
        unsigned int nb[9];
#pragma unroll
        for (int r = 0; r < 3; ++r) {
#pragma unroll
            for (int c = 0; c < 3; ++c)
                nb[r * 3 + c] = (uint32_t)(row[r] >> (8 * (i + c))) & 0xFFu;
        }
        const unsigned int burning_c = nb[4];
        const unsigned int burned_c  = (burned4 >> (8 * i)) & 0xFFu;
        const unsigned int any =
            nb[0] | nb[1] | nb[2] | nb[3] | nb[5] | nb[6] | nb[7] | nb[8];
        const size_t gi = g + i;

        float p_ignite = 0.0f;
        if (any) {
            // Slow path only where a neighbor burns (~10% of cells).
            const float V = wv[gi];
            float sw, cwv;
            __sincosf(wdir[gi] * RAD, &sw, &cwv);
            const float common =
                ph * (1.0f + p_veg[gi]) * (1.0f + p_den[gi]) * __expf(c1 * V);
            const float c2V = c2 * V;
            float prod = 1.0f;
#pragma unroll
            for (int k = 0; k < 9; ++k) {
                if (k == 4) continue;
                if (nb[k]) {
                    const float cosw = cwv * DCOS[k] + sw * DSIN[k]; // cos(theta_w)
                    const float arg  = c2V * (cosw - 1.0f) + aR * slope[gi * 9 + k];
                    const float p    = common * __expf(arg);         // p_base*p_w*p_s
                    // 1 - tanh(x) = 2 / (exp(2x) + 1)
                    const float e = __expf(2.0f * TANH_C * p);
                    prod *= 2.0f * __builtin_amdgcn_rcpf(e + 1.0f);
                }
            }
            p_ignite = 1.0f - prod;
        }

        unsigned int keep = 0u;
        if (burning_c) keep = (rand_continue[gi] < pc) ? 1u : 0u;
        unsigned int new_ignited = 0u;
        if (!burning_c && !burned_c && p_ignite > 0.0f)
            new_ignited = (rand_ignite[gi] < p_ignite) ? 1u : 0u;

        pigf[i] = p_ignite;
        nbnf[i] = (float)(new_ignited | keep);
        nbdf[i] = (float)(burned_c | (burning_c & (keep ^ 1u)));
    }

    // 16B-aligned vector stores: one b128 per output plane per thread.
    *(float4*)(out0 + g) = pig;
    *(float4*)(out1 + g) = nbn;
    *(float4*)(out2 + g) = nbd;
}

extern "C" void kernel_launch(void* const* d_in, const int* in_sizes, int n_in,
                              void* d_out, int out_size, void* d_ws, size_t ws_size,
                              hipStream_t stream) {
    (void)in_sizes; (void)n_in; (void)out_size; (void)d_ws; (void)ws_size;
    const float* p_veg = (const float*)d_in[0];
    const float* p_den = (const float*)d_in[1];
    const float* wv    = (const float*)d_in[2];
    const float* wdir  = (const float*)d_in[3];
    const float* slope = (const float*)d_in[4];
    const float* a_p   = (const float*)d_in[5];
    const float* ph_p  = (const float*)d_in[6];
    const float* c1_p  = (const float*)d_in[7];
    const float* c2_p  = (const float*)d_in[8];
    const float* pc_p  = (const float*)d_in[9];
    const float* ri    = (const float*)d_in[10];
    const float* rc    = (const float*)d_in[11];
    const unsigned char* burning = (const unsigned char*)d_in[12];
    const unsigned char* burned  = (const unsigned char*)d_in[13];
    float* out = (float*)d_out;
    const size_t plane = (size_t)HH * WW;

    dim3 grid(WW / TX, HH / TY);
    dim3 block(256);
    wildfire_step_kernel<<<grid, block, 0, stream>>>(
        p_veg, p_den, wv, wdir, slope,
        a_p, ph_p, c1_p, c2_p, pc_p,
        ri, rc, burning, burned,
        out, out + plane, out + 2 * plane);
}